// RelationAttention_87205015978176
// MI455X (gfx1250) — compile-verified
//
#include <hip/hip_runtime.h>

typedef __attribute__((ext_vector_type(2))) float v2f;
typedef __attribute__((ext_vector_type(8))) float v8f;

#define N_NODES 8192
#define F_IN    256
#define F_OUT   64
#define NEG_SLOPE 0.01f

// ---------------------------------------------------------------------------
// Kernel 1: Wh = h (8192x256) @ W (256x64), f32 WMMA 16x16x4.
// One wave per 16x16 output tile. grid = (512, 4), block = 32.
// ---------------------------------------------------------------------------
__global__ __launch_bounds__(32)
void wh_gemm_kernel(const float* __restrict__ h,
                    const float* __restrict__ W,
                    float* __restrict__ Wh) {
    const int lane = threadIdx.x & 31;
    const int mh   = lane & 15;       // m for A, n for B/C
    const int kh   = lane >> 4;       // K half-select (0 or 1)
    const int m0   = blockIdx.x * 16;
    const int n0   = blockIdx.y * 16;

    v8f acc = {};
    const float* arow = h + (size_t)(m0 + mh) * F_IN;
    for (int k = 0; k < F_IN; k += 4) {
        const int ka = k + 2 * kh;
        v2f a, b;
        a.x = arow[ka];                       // A[m][ka], A[m][ka+1] -> b64 load
        a.y = arow[ka + 1];
        b.x = W[(size_t)ka       * F_OUT + n0 + mh];
        b.y = W[(size_t)(ka + 1) * F_OUT + n0 + mh];
        acc = __builtin_amdgcn_wmma_f32_16x16x4_f32(
            /*neg_a=*/false, a, /*neg_b=*/false, b,
            /*c_mod=*/(short)0, acc, /*reuse_a=*/false, /*reuse_b=*/false);
    }
    #pragma unroll
    for (int r = 0; r < 8; ++r)
        Wh[(size_t)(m0 + r + 8 * kh) * F_OUT + n0 + mh] = acc[r];
}

// ---------------------------------------------------------------------------
// Kernel 2: f1 = Wh @ a[:64], f2 = Wh @ a[64:]. One thread per row (tiny).
// ---------------------------------------------------------------------------
__global__ __launch_bounds__(256)
void fvec_kernel(const float* __restrict__ Wh,
                 const float* __restrict__ a,
                 float* __restrict__ f1,
                 float* __restrict__ f2) {
    const int row = blockIdx.x * blockDim.x + threadIdx.x;
    if (row >= N_NODES) return;
    const float* wr = Wh + (size_t)row * F_OUT;
    float s1 = 0.f, s2 = 0.f;
    #pragma unroll
    for (int j = 0; j < F_OUT; ++j) {
        const float v = wr[j];
        s1 = fmaf(v, a[j], s1);
        s2 = fmaf(v, a[F_OUT + j], s2);
    }
    f1[row] = s1;
    f2[row] = s2;
}

// ---------------------------------------------------------------------------
// Kernel 3: fused leaky_relu(f1[i]+f2[j]) + adjacency mask + row softmax.
// One block per row; masked logits staged in LDS (32 KB) so adj is read once
// (non-temporal: read-once stream, don't pollute L2) and attention written
// once (default RT: kernel 4 re-reads it, any L2 residue is free BW).
// block = 512 (16 wave32s).
// ---------------------------------------------------------------------------
__global__ __launch_bounds__(512)
void attention_softmax_kernel(const float* __restrict__ adj,
                              const float* __restrict__ f1,
                              const float* __restrict__ f2,
                              float* __restrict__ att) {
    __shared__ float row_e[N_NODES];      // 32 KB
    __shared__ float smax[16];
    __shared__ float ssum[16];
    __shared__ float bmax, bsum;

    const int row  = blockIdx.x;
    const int tid  = threadIdx.x;
    const int lane = tid & 31;
    const int wid  = tid >> 5;
    const float NEG_INF = -__builtin_inff();

    const float fi = f1[row];
    const float* arow = adj + (size_t)row * N_NODES;

    // Pass 1: logits + mask into LDS, track running max.
    float lmax = NEG_INF;
    for (int j = tid; j < N_NODES; j += 512) {
        const float aj = __builtin_nontemporal_load(&arow[j]);   // TH=NT stream
        const float x = fi + f2[j];
        float e = (x > 0.f) ? x : NEG_SLOPE * x;
        e = (aj > 0.f) ? e : NEG_INF;
        row_e[j] = e;
        lmax = fmaxf(lmax, e);
    }
    #pragma unroll
    for (int off = 16; off > 0; off >>= 1)
        lmax = fmaxf(lmax, __shfl_xor(lmax, off, 32));
    if (lane == 0) smax[wid] = lmax;
    __syncthreads();
    if (wid == 0) {
        float v = smax[lane & 15];
        #pragma unroll
        for (int off = 8; off > 0; off >>= 1)
            v = fmaxf(v, __shfl_xor(v, off, 32));
        if (lane == 0) bmax = v;
    }
    __syncthreads();
    const float rmax = bmax;

    // Pass 2: exponentiate in LDS, accumulate sum.
    float lsum = 0.f;
    for (int j = tid; j < N_NODES; j += 512) {
        const float e = row_e[j];
        const float p = (e == NEG_INF) ? 0.f : __expf(e - rmax);
        row_e[j] = p;
        lsum += p;
    }
    #pragma unroll
    for (int off = 16; off > 0; off >>= 1)
        lsum += __shfl_xor(lsum, off, 32);
    if (lane == 0) ssum[wid] = lsum;
    __syncthreads();
    if (wid == 0) {
        float v = ssum[lane & 15];
        #pragma unroll
        for (int off = 8; off > 0; off >>= 1)
            v += __shfl_xor(v, off, 32);
        if (lane == 0) bsum = v;
    }
    __syncthreads();
    const float inv = 1.0f / bsum;

    // Pass 3: normalized write-out (the only HBM write of attention).
    float* orow = att + (size_t)row * N_NODES;
    for (int j = tid; j < N_NODES; j += 512)
        orow[j] = row_e[j] * inv;
}

// ---------------------------------------------------------------------------
// Kernel 4: h_prime = attention (8192x8192) @ Wh (8192x64), f32 WMMA 16x16x4.
// One block per 16-row strip of attention; 4 waves split K=8192; each wave
// keeps 4 accumulators (full N=64) so the attention strip is read once
// (non-temporal: 268 MB > L2, protect L2-resident Wh B-operand).
// Partials combined via LDS. grid = 512, block = 128.
// ---------------------------------------------------------------------------
__global__ __launch_bounds__(128)
void hprime_gemm_kernel(const float* __restrict__ att,
                        const float* __restrict__ Wh,
                        float* __restrict__ hp) {
    __shared__ float partial[4][4][8][32];   // [wave][n-tile][r][lane] = 16 KB

    const int tid  = threadIdx.x;
    const int lane = tid & 31;
    const int wave = tid >> 5;
    const int mh   = lane & 15;
    const int kh   = lane >> 4;
    const int m0   = blockIdx.x * 16;

    v8f acc[4] = {v8f{}, v8f{}, v8f{}, v8f{}};

    const int KSLICE = N_NODES / 4;          // 2048 per wave
    const int kbeg   = wave * KSLICE;
    const float* arow = att + (size_t)(m0 + mh) * N_NODES;

    for (int k = kbeg; k < kbeg + KSLICE; k += 4) {
        const int ka = k + 2 * kh;
        v2f a;
        a.x = __builtin_nontemporal_load(&arow[ka]);     // read-once stream
        a.y = __builtin_nontemporal_load(&arow[ka + 1]);
        const float* w0 = Wh + (size_t)ka * F_OUT;       // L2-hot B operand
        #pragma unroll
        for (int t = 0; t < 4; ++t) {
            v2f b;
            b.x = w0[t * 16 + mh];
            b.y = w0[F_OUT + t * 16 + mh];
            acc[t] = __builtin_amdgcn_wmma_f32_16x16x4_f32(
                false, a, false, b, (short)0, acc[t], false, false);
        }
    }

    #pragma unroll
    for (int t = 0; t < 4; ++t)
        #pragma unroll
        for (int r = 0; r < 8; ++r)
            partial[wave][t][r][lane] = acc[t][r];
    __syncthreads();

    if (wave == 0) {
        #pragma unroll
        for (int t = 0; t < 4; ++t)
            #pragma unroll
            for (int r = 0; r < 8; ++r) {
                const float s = partial[0][t][r][lane] + partial[1][t][r][lane]
                              + partial[2][t][r][lane] + partial[3][t][r][lane];
                hp[(size_t)(m0 + r + 8 * kh) * F_OUT + t * 16 + mh] = s;
            }
    }
}

// ---------------------------------------------------------------------------
// Launch: Wh -> (f1,f2) -> softmax(attention) -> h_prime.
// d_out = [h_prime (8192*64) | attention (8192*8192)], f32.
// d_ws  = [Wh 2 MB | f1 32 KB | f2 32 KB].
// ---------------------------------------------------------------------------
extern "C" void kernel_launch(void* const* d_in, const int* in_sizes, int n_in,
                              void* d_out, int out_size, void* d_ws, size_t ws_size,
                              hipStream_t stream) {
    const float* h   = (const float*)d_in[0];
    const float* adj = (const float*)d_in[1];
    const float* W   = (const float*)d_in[2];
    const float* a   = (const float*)d_in[3];
    // d_in[4] = rel scalar, unused

    float* out     = (float*)d_out;
    float* h_prime = out;                                // 8192*64
    float* att     = out + (size_t)N_NODES * F_OUT;      // 8192*8192

    float* Wh = (float*)d_ws;                            // 8192*64
    float* f1 = Wh + (size_t)N_NODES * F_OUT;            // 8192
    float* f2 = f1 + N_NODES;                            // 8192

    wh_gemm_kernel<<<dim3(N_NODES / 16, F_OUT / 16), 32, 0, stream>>>(h, W, Wh);
    fvec_kernel<<<dim3(N_NODES / 256), 256, 0, stream>>>(Wh, a, f1, f2);
    attention_softmax_kernel<<<dim3(N_NODES), 512, 0, stream>>>(adj, f1, f2, att);
    hprime_gemm_kernel<<<dim3(N_NODES / 16), 128, 0, stream>>>(att, Wh, h_prime);
}